// CoreNet_27118423507746
// MI455X (gfx1250) — compile-verified
//
#include <hip/hip_runtime.h>
#include <stdint.h>

// ---------------------------------------------------------------------------
// Problem: scores[i,j] = sa[i] + sb[j] + sum_d |e[i,d]-e[j,d]| * wc[d] + b
//   e: (1024,256) f32, W: (768,1) f32 [wa|wb|wc], b: (1,) f32
//   out: (1024,1024) f32 row-major
//
// The pairwise weighted-L1 term is not bilinear -> VALU-bound (2 VALU per
// (i,j,d)).  The separable sa/sb matvecs are exact-f32 WMMA.  Tile staging
// uses the CDNA5 async global->LDS engine, double-buffered on ASYNCcnt.
// ---------------------------------------------------------------------------

typedef __attribute__((ext_vector_type(2))) float v2f;
typedef __attribute__((ext_vector_type(8))) float v8f;

#define N_ROWS 1024
#define DIM    256
#define TI     64      // output tile (both i and j)
#define KC     32      // k chunk staged in LDS (double-buffered)
#define NCHUNK (DIM / KC)
#define OPS_PER_CHUNK 16   // async b32 ops issued per thread per chunk (8 A + 8 B)

// ---------------------------------------------------------------------------
// Kernel 1: sa = e @ wa, sb = e @ wb via V_WMMA_F32_16X16X4_F32 (exact f32).
// One wave per block; block owns 16 rows of e.  B has wa in column 0, wb in
// column 1, zeros elsewhere (branchless: lanes m>=2 load valid memory and
// select 0, keeping EXEC all-ones around the WMMAs).
//
// A layout (ISA 7.12.2, 32-bit A 16x4): lanes 0-15 hold M=0..15 with
//   VGPR0=K0, VGPR1=K1; lanes 16-31 hold K2/K3.
// B layout (4x16, rows striped across lanes): lanes 0-15 = rows K0/K1,
//   lanes 16-31 = rows K2/K3 (mirror of A).
// C/D layout: lane n (mod 16) = column; VGPR r holds M=r (lanes 0-15) and
//   M=r+8 (lanes 16-31).
// ---------------------------------------------------------------------------
__global__ __launch_bounds__(32)
void sab_wmma_kernel(const float* __restrict__ e,
                     const float* __restrict__ W,
                     float* __restrict__ sab /* [2][1024] in workspace */) {
    const int lane = threadIdx.x;      // 0..31
    const int i0   = blockIdx.x * 16;  // row block
    const int m    = lane & 15;        // A: row M; B: column N
    const int hi   = lane >> 4;        // 0 -> K{0,1}, 1 -> K{2,3}

#if __has_builtin(__builtin_amdgcn_wmma_f32_16x16x4_f32)
    const float  sel = (m < 2) ? 1.0f : 0.0f;        // zero columns n>=2
    const float* wp  = W + (m & 1) * DIM;            // n==0 -> wa, n==1 -> wb
    const float* ap  = e + (i0 + m) * DIM;

    v8f c = {};
    for (int k = 0; k < DIM; k += 4) {
        const int ka = k + hi * 2;
        v2f a;
        a.x = ap[ka + 0];
        a.y = ap[ka + 1];
        v2f bm;
        bm.x = wp[ka + 0] * sel;
        bm.y = wp[ka + 1] * sel;
        // (neg_a, A, neg_b, B, c_mod, C, reuse_a, reuse_b)
        c = __builtin_amdgcn_wmma_f32_16x16x4_f32(false, a, false, bm,
                                                  (short)0, c, false, false);
    }
    if (m < 2) {
        float* dst = sab + m * N_ROWS;     // row 0: sa, row 1: sb
        #pragma unroll
        for (int r = 0; r < 8; ++r)
            dst[i0 + 8 * hi + r] = c[r];
    }
#else
    // Scalar fallback: lanes 0..15 -> sa rows, 16..31 -> sb rows.
    const float* wcol = W + hi * DIM;
    float s = 0.0f;
    for (int k = 0; k < DIM; ++k)
        s = __builtin_fmaf(e[(i0 + m) * DIM + k], wcol[k], s);
    sab[hi * N_ROWS + i0 + m] = s;
#endif
}

// ---------------------------------------------------------------------------
// Async staging helper: copy chunk `cb` of the i-tile and j-tile into LDS
// buffer `buf`, TRANSPOSED to [k][row] (per-lane LDS destination does the
// transpose in flight).  16 async b32 ops per thread.
// ---------------------------------------------------------------------------
__device__ __forceinline__
void stage_chunk(uint64_t eAddr, uint32_t ldsA, uint32_t ldsB,
                 uint32_t iBase, uint32_t jBase, uint32_t tid,
                 int cb, int buf) {
    const uint32_t bufOff = (uint32_t)buf * (KC * TI * 4u);
    const uint32_t kOff   = (uint32_t)cb * KC;
    #pragma unroll
    for (int t = 0; t < 8; ++t) {
        const uint32_t flat = (uint32_t)(t * 256) + tid;   // 0..2047
        const uint32_t r    = flat >> 5;                   // tile row 0..63
        const uint32_t kk   = flat & 31;                   // k within chunk
        const uint32_t goffA = ((iBase + r) * DIM + kOff + kk) * 4u;
        const uint32_t goffB = ((jBase + r) * DIM + kOff + kk) * 4u;
        const uint32_t la = ldsA + bufOff + (kk * TI + r) * 4u;
        const uint32_t lb = ldsB + bufOff + (kk * TI + r) * 4u;
        asm volatile("global_load_async_to_lds_b32 %0, %1, %2"
                     :: "v"(la), "v"(goffA), "s"(eAddr) : "memory");
        asm volatile("global_load_async_to_lds_b32 %0, %1, %2"
                     :: "v"(lb), "v"(goffB), "s"(eAddr) : "memory");
    }
}

// ---------------------------------------------------------------------------
// Kernel 2: pairwise weighted-L1 term, fused with sa/sb/b epilogue.
// 64x64 output tile per block, 256 threads, 4x4 register micro-tile.
// Double-buffered async staging: issue chunk n+1, s_wait_asynccnt <= 16
// (in-order completion) guarantees chunk n has landed, then barrier.
// ---------------------------------------------------------------------------
__global__ __launch_bounds__(256)
void pairwise_kernel(const float* __restrict__ e,
                     const float* __restrict__ W,
                     const float* __restrict__ bias,
                     const float* __restrict__ sab,
                     float* __restrict__ out) {
    __shared__ float sA[2][KC][TI];   // e_i tiles, [buf][k][i]  (16 KB)
    __shared__ float sB[2][KC][TI];   // e_j tiles, [buf][k][j]  (16 KB)
    __shared__ float sw[DIM];         // wc                       (1 KB)

    const uint32_t tid   = threadIdx.x;
    const uint32_t iBase = blockIdx.y * TI;
    const uint32_t jBase = blockIdx.x * TI;
    const int      tx    = tid & 15;    // j group (4 cols each)
    const int      ty    = tid >> 4;    // i group (4 rows each)

    sw[tid] = W[2 * DIM + tid];         // wc = W[512:768]; 256 threads cover it

    const uint32_t ldsA  = (uint32_t)(uintptr_t)&sA[0][0][0];
    const uint32_t ldsB  = (uint32_t)(uintptr_t)&sB[0][0][0];
    const uint64_t eAddr = (uint64_t)(uintptr_t)e;

    float acc[4][4] = {};

    stage_chunk(eAddr, ldsA, ldsB, iBase, jBase, tid, /*cb=*/0, /*buf=*/0);

    for (int cb = 0; cb < NCHUNK; ++cb) {
        const int cur = cb & 1;
        if (cb + 1 < NCHUNK) {
            // Prefetch next chunk into the other buffer (last read before the
            // previous iteration's trailing barrier -> WAR safe).
            stage_chunk(eAddr, ldsA, ldsB, iBase, jBase, tid, cb + 1, cur ^ 1);
            // 16 ops of chunk cb+1 outstanding; ASYNCcnt completes in order,
            // so <=16 means all of chunk cb's copies have landed in LDS.
            asm volatile("s_wait_asynccnt 0x10" ::: "memory");
        } else {
            asm volatile("s_wait_asynccnt 0x0" ::: "memory");
        }
        __syncthreads();

        // ---- compute: 4x4 micro-tile, 2 VALU per (i,j,d) ------------------
        const int wbase = cb * KC;
        #pragma unroll 8
        for (int k = 0; k < KC; ++k) {
            const float  wv = sw[wbase + k];
            const float4 av = *(const float4*)&sA[cur][k][ty * 4];  // ds_load_b128
            const float4 bv = *(const float4*)&sB[cur][k][tx * 4];  // ds_load_b128
            const float ar[4] = {av.x, av.y, av.z, av.w};
            const float br[4] = {bv.x, bv.y, bv.z, bv.w};
            #pragma unroll
            for (int ii = 0; ii < 4; ++ii)
                #pragma unroll
                for (int jj = 0; jj < 4; ++jj)
                    acc[ii][jj] = __builtin_fmaf(
                        __builtin_fabsf(ar[ii] - br[jj]), wv, acc[ii][jj]);
        }
        __syncthreads();
    }

    // ---- epilogue: + sa[i] + sb[j] + b, b128 stores ----------------------
    const float bb = bias[0];
    float sbj[4];
    #pragma unroll
    for (int jj = 0; jj < 4; ++jj)
        sbj[jj] = sab[N_ROWS + jBase + tx * 4 + jj];

    #pragma unroll
    for (int ii = 0; ii < 4; ++ii) {
        const int   i   = iBase + ty * 4 + ii;
        const float sai = sab[i] + bb;
        float4 o;
        o.x = acc[ii][0] + sai + sbj[0];
        o.y = acc[ii][1] + sai + sbj[1];
        o.z = acc[ii][2] + sai + sbj[2];
        o.w = acc[ii][3] + sai + sbj[3];
        *(float4*)&out[(size_t)i * N_ROWS + jBase + tx * 4] = o;
    }
}

// ---------------------------------------------------------------------------
extern "C" void kernel_launch(void* const* d_in, const int* in_sizes, int n_in,
                              void* d_out, int out_size, void* d_ws, size_t ws_size,
                              hipStream_t stream) {
    const float* e    = (const float*)d_in[0];  // (1024, 256)
    const float* W    = (const float*)d_in[1];  // (768, 1)
    const float* bias = (const float*)d_in[2];  // (1,)
    float*       out  = (float*)d_out;          // (1024, 1024)
    float*       sab  = (float*)d_ws;           // [2][1024] scratch

    sab_wmma_kernel<<<N_ROWS / 16, 32, 0, stream>>>(e, W, sab);
    pairwise_kernel<<<dim3(N_ROWS / TI, N_ROWS / TI), 256, 0, stream>>>(
        e, W, bias, sab, out);
}